// VADE_36515811950932
// MI455X (gfx1250) — compile-verified
//
#include <hip/hip_runtime.h>
#include <math.h>

typedef __attribute__((ext_vector_type(2))) float v2f;
typedef __attribute__((ext_vector_type(8))) float v8f;

#define B_TOTAL 262144
#define LOG2PI 1.8378770664093453f
#define TWOPI  6.283185307179586f

// ---------------------------------------------------------------------------
// Prep: build the fused weight matrix W[32][128] (row-major) in d_ws.
//   features: f = [z(10), z^2(10), exp(zlv)(10), 1, 0]
//   cols 0..63  : logit s_bc = D*log(theta_c) + log_lik  (bias folded in row 30)
//   cols 64..127: T'_bc = sum_d term - 2*log(theta_c)    (bias folded in row 30)
// ---------------------------------------------------------------------------
__global__ void vade_prep_kernel(const float* __restrict__ theta,
                                 const float* __restrict__ u,
                                 const float* __restrict__ lam,
                                 float* __restrict__ W) {
  int j = threadIdx.x;           // 0..127 = output column
  if (j >= 128) return;
  int c = j & 63;
  bool isT = (j >= 64);
  float acc = 0.0f;
  for (int d = 0; d < 10; ++d) {
    float l   = lam[d * 64 + c];
    float uu  = u[d * 64 + c];
    float inv = 1.0f / l;
    if (!isT) {
      W[d * 128 + j]        = uu * inv;        // z coeff
      W[(10 + d) * 128 + j] = -0.5f * inv;     // z^2 coeff
      W[(20 + d) * 128 + j] = 0.0f;            // exp(zlv) unused for logits
      acc += -0.5f * logf(TWOPI * l) - 0.5f * uu * uu * inv;
    } else {
      W[d * 128 + j]        = -2.0f * uu * inv; // z coeff of diff2/lam
      W[(10 + d) * 128 + j] = inv;              // z^2 coeff
      W[(20 + d) * 128 + j] = inv;              // exp(zlv)/lam coeff
      acc += logf(l) + uu * uu * inv;
    }
  }
  float lt = logf(theta[c]);
  W[30 * 128 + j] = isT ? (100.0f * LOG2PI + acc - 2.0f * lt)  // D^2*log2pi + const - 2logtheta
                        : (10.0f * lt + acc);                  // D*logtheta + const
  W[31 * 128 + j] = 0.0f;                                      // padding feature
}

// ---------------------------------------------------------------------------
// Main: [B,32] x [32,128] GEMM on v_wmma_f32_16x16x4_f32 + fused epilogue.
// 256 threads = 8 wave32; each wave owns one 16-row tile (128 rows/block).
// W is staged in LDS pre-swizzled into WMMA B-fragment order:
//   Wp[kk][half][col][2] holds the (K=4kk+2*half, K+1) pair contiguously,
// so each B operand is one aligned ds_load_b64 straight into an even VGPR
// pair (no v_mov repacking). half stride 288 == 32 (mod 64) keeps the two
// half-wave groups on disjoint bank sets -> conflict-free.
// ---------------------------------------------------------------------------
__global__ __launch_bounds__(256)
void vade_main_kernel(const float* __restrict__ z_mean,
                      const float* __restrict__ z_log_var,
                      const float* __restrict__ Wg,
                      float* __restrict__ out) {
  __shared__ float Wp[8 * 576];        // 8 k-chunks * (2 halves * 288)
  // Per-wave A (feature) tiles, row stride 36: 36*l mod 64 distinct for l<16.
  __shared__ float F[8 * 16 * 36];
  __shared__ float ZS[8 * 16];         // per-row sum of z_log_var

  const int tid = threadIdx.x;

  // Cooperative load of W, scattering into B-fragment order (coalesced reads).
#pragma unroll
  for (int i = 0; i < 16; ++i) {
    int idx = tid + i * 256;
    int k = idx >> 7;                  // feature row 0..31
    int c = idx & 127;                 // output column
    int kk = k >> 2;                   // k-chunk
    int r  = k & 3;                    // row within chunk
    Wp[kk * 576 + (r >> 1) * 288 + c * 2 + (r & 1)] = Wg[idx];
  }

  const int wave = tid >> 5;
  const int lane = tid & 31;
  const int li   = lane & 15;
  const int hi   = (lane & 16) ? 1 : 0;
  const long base = ((long)blockIdx.x * 8 + wave) * 16;

  // ---- Feature stage: lanes 0-15 build z/z^2, lanes 16-31 build exp(zlv). ----
  float* fr = &F[wave * 576 + li * 36];
  if (!hi) {
    const float2* zp = (const float2*)z_mean + (base + li) * 5;  // 8B aligned
#pragma unroll
    for (int i = 0; i < 5; ++i) {
      float2 v = zp[i];
      fr[2 * i]      = v.x;       fr[2 * i + 1]  = v.y;
      fr[10 + 2 * i] = v.x * v.x; fr[11 + 2 * i] = v.y * v.y;
    }
  } else {
    const float2* lp = (const float2*)z_log_var + (base + li) * 5;
    float s = 0.0f;
#pragma unroll
    for (int i = 0; i < 5; ++i) {
      float2 v = lp[i];
      fr[20 + 2 * i] = __expf(v.x);
      fr[21 + 2 * i] = __expf(v.y);
      s += v.x + v.y;
    }
    fr[30] = 1.0f;   // bias feature
    fr[31] = 0.0f;   // pad
    ZS[wave * 16 + li] = s;
  }
  __syncthreads();

  // ---- Load A chunks (16x4 fp32 layout: lanes 0-15 K0,K1; lanes 16-31 K2,K3). ----
  const int koff = hi ? 2 : 0;
  const float* frA = &F[wave * 576 + li * 36 + koff];  // 8B aligned
  v2f a[8];
#pragma unroll
  for (int kk = 0; kk < 8; ++kk) {
    a[kk] = *(const v2f*)(frA + 4 * kk);               // ds_load_b64, conflict-free
  }

  const v8f vzero = {0.f, 0.f, 0.f, 0.f, 0.f, 0.f, 0.f, 0.f};
  v8f acc[8];
#pragma unroll
  for (int n = 0; n < 8; ++n) acc[n] = vzero;

  // ---- GEMM: 8 k-chunks x 8 column tiles = 64 WMMAs per wave. ----
#pragma unroll
  for (int kk = 0; kk < 8; ++kk) {
    const float* wp = &Wp[kk * 576 + hi * 288 + li * 2];
#pragma unroll
    for (int n = 0; n < 8; ++n) {
      v2f b = *(const v2f*)(wp + 32 * n);              // one ds_load_b64 per fragment
      acc[n] = __builtin_amdgcn_wmma_f32_16x16x4_f32(
          false, a[kk], false, b, (short)0, acc[n], false, false);
    }
  }

  // ---- Epilogue: p=exp(s)+1e-10; loss = sum gamma*(0.5T'+log p) - log S - 0.5*(sum zlv + D). ----
#pragma unroll
  for (int r = 0; r < 8; ++r) {
    float ps = 0.0f, pv = 0.0f;
#pragma unroll
    for (int n = 0; n < 4; ++n) {
      float s = acc[n][r];        // logits live in column tiles 0..3
      float t = acc[n + 4][r];    // T' lives in column tiles 4..7 (same lane/col)
      float p = __expf(s) + 1e-10f;
      ps += p;
      pv += p * (0.5f * t + __logf(p));
    }
    // Reduce across the 16-lane half-group (masks <16 stay in-group on wave32).
#pragma unroll
    for (int m = 1; m < 16; m <<= 1) {
      ps += __shfl_xor(ps, m, 32);
      pv += __shfl_xor(pv, m, 32);
    }
    if (li == r) {
      float zs = ZS[wave * 16 + hi * 8 + r];
      out[base + hi * 8 + r] = pv / ps - __logf(ps) - 0.5f * (zs + 10.0f);
    }
  }
}

// ---------------------------------------------------------------------------
extern "C" void kernel_launch(void* const* d_in, const int* in_sizes, int n_in,
                              void* d_out, int out_size, void* d_ws, size_t ws_size,
                              hipStream_t stream) {
  (void)in_sizes; (void)n_in; (void)out_size; (void)ws_size;
  const float* z_mean    = (const float*)d_in[0];
  const float* z_log_var = (const float*)d_in[1];
  const float* theta_p   = (const float*)d_in[2];
  const float* u_p       = (const float*)d_in[3];
  const float* lambda_p  = (const float*)d_in[4];
  float* W = (float*)d_ws;   // 32*128 floats = 16 KB scratch

  vade_prep_kernel<<<1, 128, 0, stream>>>(theta_p, u_p, lambda_p, W);
  vade_main_kernel<<<B_TOTAL / 128, 256, 0, stream>>>(z_mean, z_log_var, W,
                                                      (float*)d_out);
}